// DeformableTransformerDecoderLayer_82858509074987
// MI455X (gfx1250) — compile-verified
//
#include <hip/hip_runtime.h>

// ---------------------------------------------------------------------------
// DeformableTransformerDecoderLayer for MI455X (gfx1250, wave32).
// - All GEMMs through V_WMMA_F32_16X16X4_F32 (fp32 matches reference numerics;
//   workload is bandwidth-bound: ~66 GFLOP fp32 vs 23.3 TB/s HBM).
// - GEMM tile staging uses GLOBAL_LOAD_ASYNC_TO_LDS_B128 with double-buffered
//   LDS (ASYNCcnt-tracked), overlapping global traffic with WMMA compute.
// ---------------------------------------------------------------------------

typedef float v2f __attribute__((ext_vector_type(2)));
typedef float v8f __attribute__((ext_vector_type(8)));
typedef int   v4i_t __attribute__((vector_size(4 * sizeof(int))));

#define D_    256
#define DFF_  1024
#define H_    8
#define NK_   18
#define NQ_   100
#define BS_   8
#define LIN_  13294
#define T_    (NQ_ * BS_ * NK_)   // 14400 tokens

// -------------------- CDNA5 async copy / wait helpers ----------------------
__device__ __forceinline__ void async_b128(const float* g, float* l)
{
#if __has_builtin(__builtin_amdgcn_global_load_async_to_lds_b128)
  v4i_t* gg = (v4i_t*)g;   // drop const / retype in generic AS
  v4i_t* ll = (v4i_t*)l;
  __builtin_amdgcn_global_load_async_to_lds_b128(
      (__attribute__((address_space(1))) v4i_t*)gg,
      (__attribute__((address_space(3))) v4i_t*)ll, 0, 0);
#else
  unsigned laddr = (unsigned)(size_t)(__attribute__((address_space(3))) float*)l;
  unsigned long long gaddr = (unsigned long long)(size_t)g;
  asm volatile("global_load_async_to_lds_b128 %0, %1, off"
               :: "v"(laddr), "v"(gaddr) : "memory");
#endif
}

template <int N>
__device__ __forceinline__ void wait_asynccnt()
{
#if __has_builtin(__builtin_amdgcn_s_wait_asynccnt)
  __builtin_amdgcn_s_wait_asynccnt(N);
#else
  asm volatile("s_wait_asynccnt %0" :: "i"(N) : "memory");
#endif
}

// =============================== GEMM ======================================
// Y[M,N] = act(X[M,K] @ W[N,K]^T + bias[N] (+ res[M,N]))
// 64x64 block tile, 4 waves each owning a 32x32 tile (2x2 WMMA 16x16 f32
// accumulators). K staged in BK=16 chunks through double-buffered LDS filled
// by async B128 loads. Row stride 20 floats (80B): 16B-aligned for B128 LDS
// writes and bank-conflict-free for fragment reads.
#define BM  64
#define BN  64
#define BK  16
#define LDW (BK + 4)   // 20 floats

__global__ __launch_bounds__(128) void gemm_f32_wmma(
    const float* __restrict__ X, const float* __restrict__ W,
    const float* __restrict__ bias, const float* __restrict__ res,
    float* __restrict__ Y, int M, int N, int K, int relu)
{
  __shared__ float Xs[2][BM][LDW];
  __shared__ float Ws[2][BN][LDW];
  const int tid  = threadIdx.x;
  const int lane = tid & 31;
  const int wave = tid >> 5;
  const int half = lane >> 4;   // 0: lanes 0-15, 1: lanes 16-31
  const int l16  = lane & 15;
  const int bm = blockIdx.y * BM;
  const int bn = blockIdx.x * BN;
  const int wy = (wave >> 1) * 32;   // wave's row offset in block tile
  const int wx = (wave & 1) * 32;    // wave's col offset in block tile

  v8f acc[2][2] = {};

  // 4 WMMA k-steps consuming LDS buffer cb.
  auto compute = [&](int cb) {
#pragma unroll
    for (int kk = 0; kk < BK; kk += 4) {
      // A 16x4: lanes 0-15 -> K {kk,kk+1}; lanes 16-31 -> K {kk+2,kk+3}.
      // B 4x16: same K split, N across lanes.
      v2f a[2], b[2];
#pragma unroll
      for (int i = 0; i < 2; ++i) {
        int row = wy + i * 16 + l16;
        a[i].x = Xs[cb][row][kk + 2 * half + 0];
        a[i].y = Xs[cb][row][kk + 2 * half + 1];
        int col = wx + i * 16 + l16;
        b[i].x = Ws[cb][col][kk + 2 * half + 0];
        b[i].y = Ws[cb][col][kk + 2 * half + 1];
      }
#pragma unroll
      for (int i = 0; i < 2; ++i)
#pragma unroll
        for (int jj = 0; jj < 2; ++jj)
          acc[i][jj] = __builtin_amdgcn_wmma_f32_16x16x4_f32(
              false, a[i], false, b[jj], (short)0, acc[i][jj], false, false);
    }
  };

  // Async stage: 64x16 X tile + 64x16 W tile; 256 float4 slots each,
  // 2 per thread per matrix -> 4 async loads / thread / stage.
  auto stage_async = [&](int s, int cb) {
    int k0 = s * BK;
#pragma unroll
    for (int i = 0; i < 2; ++i) {
      int slot = tid + i * 128;
      int r = slot >> 2;
      int c = (slot & 3) << 2;
      async_b128(X + (size_t)(bm + r) * K + k0 + c, &Xs[cb][r][c]);
      async_b128(W + (size_t)(bn + r) * K + k0 + c, &Ws[cb][r][c]);
    }
  };

  // Guarded synchronous stage for M/N-edge blocks (zero-fill OOB rows).
  auto stage_sync = [&](int s) {
    int k0 = s * BK;
    for (int i = tid; i < BM * (BK / 4); i += 128) {
      int r = i >> 2;
      int c = (i & 3) << 2;
      float4 vx = make_float4(0.f, 0.f, 0.f, 0.f);
      float4 vw = make_float4(0.f, 0.f, 0.f, 0.f);
      int gr = bm + r;
      if (gr < M) vx = *(const float4*)(X + (size_t)gr * K + k0 + c);
      int gn = bn + r;
      if (gn < N) vw = *(const float4*)(W + (size_t)gn * K + k0 + c);
      Xs[0][r][c + 0] = vx.x; Xs[0][r][c + 1] = vx.y;
      Xs[0][r][c + 2] = vx.z; Xs[0][r][c + 3] = vx.w;
      Ws[0][r][c + 0] = vw.x; Ws[0][r][c + 1] = vw.y;
      Ws[0][r][c + 2] = vw.z; Ws[0][r][c + 3] = vw.w;
    }
  };

  const int nstage = K / BK;
  const bool interior = (bm + BM <= M) && (bn + BN <= N);   // block-uniform

  if (interior) {
    stage_async(0, 0);
    for (int s = 0; s < nstage; ++s) {
      int cb = s & 1;
      if (s + 1 < nstage) {
        stage_async(s + 1, cb ^ 1);       // prefetch next stage
        wait_asynccnt<4>();               // in-order: current stage done
      } else {
        wait_asynccnt<0>();
      }
      __syncthreads();
      compute(cb);
      __syncthreads();                    // before next prefetch reuses LDS
    }
  } else {
    for (int s = 0; s < nstage; ++s) {
      stage_sync(s);
      __syncthreads();
      compute(0);
      __syncthreads();
    }
  }

  // Epilogue: C/D layout -> VGPR r holds row r (lanes 0-15) / row r+8 (16-31).
#pragma unroll
  for (int i = 0; i < 2; ++i) {
#pragma unroll
    for (int jj = 0; jj < 2; ++jj) {
      int col = bn + wx + jj * 16 + l16;
      if (col >= N) continue;
      float bv = bias ? bias[col] : 0.f;
#pragma unroll
      for (int r = 0; r < 8; ++r) {
        int row = bm + wy + i * 16 + half * 8 + r;
        if (row >= M) continue;
        float v = acc[i][jj][r] + bv;
        if (res) v += res[(size_t)row * N + col];
        if (relu) v = fmaxf(v, 0.f);
        Y[(size_t)row * N + col] = v;
      }
    }
  }
}

// ============================ Attention ====================================
// Generic MHA core over packed qkv[t][768] (q|k|v of 256 each, 8 heads x 32).
// Token index for (seq s, batch j): t = s*ss + j*sj.
//   across: S=100, J=144, ss=144, sj=1
//   within: S=18,  J=800, ss=1,   sj=18, maskAdd=1e-8
#define SMAX 100

__global__ __launch_bounds__(128) void attn_softmax(
    const float* __restrict__ qkv, float* __restrict__ out,
    int S, int ss, int sj, float maskAdd)
{
  __shared__ float Ks[SMAX * 33];   // stride 33: conflict-free dot + AV loops
  __shared__ float Vs[SMAX * 33];
  __shared__ float Ps[4][SMAX];
  __shared__ float Qs[4][32];
  const int tid = threadIdx.x, wave = tid >> 5, lane = tid & 31;
  const int j = blockIdx.x, h = blockIdx.y;
  const float scale = 0.17677669529663687f;   // 1/sqrt(32)

  for (int i = tid; i < S * 32; i += 128) {
    int s = i >> 5, d = i & 31;
    size_t t = (size_t)s * ss + (size_t)j * sj;
    Ks[s * 33 + d] = qkv[t * 768 + 256 + h * 32 + d];
    Vs[s * 33 + d] = qkv[t * 768 + 512 + h * 32 + d];
  }
  __syncthreads();

  const int iters = (S + 3) >> 2;    // uniform loop count for barriers
  for (int it = 0; it < iters; ++it) {
    int s0 = it * 4 + wave;
    bool act = s0 < S;
    size_t t0 = act ? ((size_t)s0 * ss + (size_t)j * sj) : 0;
    Qs[wave][lane] = act ? qkv[t0 * 768 + h * 32 + lane] * scale : 0.f;
    __syncthreads();

    float sc[4];
    float m = -3.0e38f;
    const int nk = (S + 31) >> 5;
    for (int kb = 0; kb < nk; ++kb) {
      int key = kb * 32 + lane;
      float dot = -3.0e38f;
      if (key < S) {
        dot = maskAdd;
        for (int c = 0; c < 32; ++c)
          dot = fmaf(Qs[wave][c], Ks[key * 33 + c], dot);
      }
      sc[kb] = dot;
      m = fmaxf(m, dot);
    }
    for (int msk = 16; msk; msk >>= 1) m = fmaxf(m, __shfl_xor(m, msk, 32));
    float sum = 0.f;
    for (int kb = 0; kb < nk; ++kb) {
      int key = kb * 32 + lane;
      if (key < S) { float p = __expf(sc[kb] - m); Ps[wave][key] = p; sum += p; }
    }
    for (int msk = 16; msk; msk >>= 1) sum += __shfl_xor(sum, msk, 32);
    __syncthreads();

    float accv = 0.f;
    for (int key = 0; key < S; ++key)
      accv = fmaf(Ps[wave][key], Vs[key * 33 + lane], accv);
    if (act) out[t0 * 256 + h * 32 + lane] = accv / sum;
    __syncthreads();
  }
}

// ====================== MS-Deformable sampling =============================
// One block per (query lq, batch b); wave = head, lane = channel (DH=32).
__global__ __launch_bounds__(256) void msdeform_sample(
    const float* __restrict__ off, const float* __restrict__ aw,
    const float* __restrict__ ref, const float* __restrict__ value,
    float* __restrict__ out)
{
  const int blk = blockIdx.x;
  const int b  = blk % BS_;
  const int lq = blk / BS_;
  const int h    = threadIdx.x >> 5;
  const int lane = threadIdx.x & 31;
  const int q = lq / NK_, kk = lq % NK_;
  const size_t t = ((size_t)q * BS_ + b) * NK_ + kk;   // token row

  // softmax over 16 attention weights (lanes 0-15 / 16-31 hold duplicates)
  float av = aw[t * 128 + h * 16 + (lane & 15)];
  float m = av;
  for (int msk = 8; msk; msk >>= 1) m = fmaxf(m, __shfl_xor(m, msk, 32));
  float e = __expf(av - m);
  float s = e;
  for (int msk = 8; msk; msk >>= 1) s += __shfl_xor(s, msk, 32);
  av = e / s;

  const int WlA[4] = {100, 50, 25, 13};
  const int HlA[4] = {100, 50, 25, 13};
  const int STA[4] = {0, 10000, 12500, 13125};

  float accv = 0.f;
#pragma unroll
  for (int l = 0; l < 4; ++l) {
    const int   Wl = WlA[l], Hl = HlA[l], ST = STA[l];
    const float fw = (float)Wl, fh = (float)Hl;
    float rx = ref[((size_t)lq * BS_ + b) * 8 + l * 2 + 0];
    float ry = ref[((size_t)lq * BS_ + b) * 8 + l * 2 + 1];
#pragma unroll
    for (int p = 0; p < 4; ++p) {
      float ox = off[t * 256 + (((h * 4 + l) * 4 + p) * 2) + 0];
      float oy = off[t * 256 + (((h * 4 + l) * 4 + p) * 2) + 1];
      float a = __shfl(av, l * 4 + p, 32);
      // loc = ref + off/norm; x = loc.x*Wl - 0.5  ==  rx*Wl + ox - 0.5
      float x = fmaf(rx, fw, ox) - 0.5f;
      float y = fmaf(ry, fh, oy) - 0.5f;
      float x0f = floorf(x), y0f = floorf(y);
      float lx = x - x0f, ly = y - y0f;
      int x0 = (int)x0f, y0 = (int)y0f;
      float wgt[4] = {(1.f - lx) * (1.f - ly), lx * (1.f - ly),
                      (1.f - lx) * ly,         lx * ly};
      int xs[4] = {x0, x0 + 1, x0, x0 + 1};
      int ys[4] = {y0, y0, y0 + 1, y0 + 1};
#pragma unroll
      for (int c = 0; c < 4; ++c) {
        int xi = xs[c], yi = ys[c];
        if (xi >= 0 && xi < Wl && yi >= 0 && yi < Hl) {
          size_t vi = ((size_t)(ST + yi * Wl + xi) * BS_ + b) * 256 + h * 32 + lane;
          accv = fmaf(a * wgt[c], value[vi], accv);
        }
      }
    }
  }
  out[t * 256 + h * 32 + lane] = accv;
}

// ============================ LayerNorm ====================================
// One wave per token (8 dims per lane); optional fused "+= addafter" (pos).
__global__ __launch_bounds__(256) void layernorm_k(
    const float* __restrict__ in, const float* __restrict__ g,
    const float* __restrict__ bb, const float* __restrict__ addafter,
    float* __restrict__ outp, int T)
{
  const int wave = threadIdx.x >> 5, lane = threadIdx.x & 31;
  const size_t t = (size_t)blockIdx.x * 8 + wave;
  if (t >= (size_t)T) return;
  const float* row = in + t * 256;
  float v[8];
  float s = 0.f;
#pragma unroll
  for (int i = 0; i < 8; ++i) { v[i] = row[lane + 32 * i]; s += v[i]; }
  for (int msk = 16; msk; msk >>= 1) s += __shfl_xor(s, msk, 32);
  float mu = s * (1.f / 256.f);
  float var = 0.f;
#pragma unroll
  for (int i = 0; i < 8; ++i) { float d = v[i] - mu; var = fmaf(d, d, var); }
  for (int msk = 16; msk; msk >>= 1) var += __shfl_xor(var, msk, 32);
  float rs = rsqrtf(var * (1.f / 256.f) + 1e-5f);
#pragma unroll
  for (int i = 0; i < 8; ++i) {
    int d = lane + 32 * i;
    float o = (v[i] - mu) * rs * g[d] + bb[d];
    if (addafter) o += addafter[t * 256 + d];
    outp[t * 256 + d] = o;
  }
}

// ============================== Driver =====================================
extern "C" void kernel_launch(void* const* d_in, const int* in_sizes, int n_in,
                              void* d_out, int out_size, void* d_ws, size_t ws_size,
                              hipStream_t stream)
{
  const float* tgt  = (const float*)d_in[0];
  const float* pos  = (const float*)d_in[1];
  const float* refp = (const float*)d_in[2];
  const float* mem  = (const float*)d_in[3];
  const float* across_in_w  = (const float*)d_in[4];
  const float* across_in_b  = (const float*)d_in[5];
  const float* across_out_w = (const float*)d_in[6];
  const float* across_out_b = (const float*)d_in[7];
  const float* within_in_w  = (const float*)d_in[8];
  const float* within_in_b  = (const float*)d_in[9];
  const float* within_out_w = (const float*)d_in[10];
  const float* within_out_b = (const float*)d_in[11];
  const float* off_w  = (const float*)d_in[12];
  const float* off_b  = (const float*)d_in[13];
  const float* aw_w   = (const float*)d_in[14];
  const float* aw_b   = (const float*)d_in[15];
  const float* val_w  = (const float*)d_in[16];
  const float* val_b  = (const float*)d_in[17];
  const float* msout_w = (const float*)d_in[18];
  const float* msout_b = (const float*)d_in[19];
  const float* lin1_w  = (const float*)d_in[20];
  const float* lin1_b  = (const float*)d_in[21];
  const float* lin2_w  = (const float*)d_in[22];
  const float* lin2_b  = (const float*)d_in[23];
  const float* lin1p_w = (const float*)d_in[24];
  const float* lin1p_b = (const float*)d_in[25];
  const float* lin2p_w = (const float*)d_in[26];
  const float* lin2p_b = (const float*)d_in[27];
  const float* across_norm_g = (const float*)d_in[28];
  const float* across_norm_b = (const float*)d_in[29];
  const float* within_norm_g = (const float*)d_in[30];
  const float* within_norm_b = (const float*)d_in[31];
  const float* norm1_g  = (const float*)d_in[32];
  const float* norm1_b  = (const float*)d_in[33];
  const float* norm2_g  = (const float*)d_in[34];
  const float* norm2_b  = (const float*)d_in[35];
  const float* norm2p_g = (const float*)d_in[36];
  const float* norm2p_b = (const float*)d_in[37];

  float* ws = (float*)d_ws;
  const size_t T = (size_t)T_;
  float* buf_x   = ws;                    // T*256  current activations
  float* buf_a   = buf_x   + T * 256;     // T*256  attn/sample heads out
  float* buf_b   = buf_a   + T * 256;     // T*256  pre-LN
  float* buf_h   = buf_b   + T * 256;     // T*1024 qkv (768) / FFN hidden
  float* buf_off = buf_h   + T * 1024;    // T*256
  float* buf_aw  = buf_off + T * 256;     // T*128
  float* buf_val = buf_aw  + T * 128;     // (LIN*BS)*256

  auto gemm = [&](const float* X, const float* W, const float* bias,
                  const float* res, float* Y, int M, int N, int K, int relu) {
    dim3 grid((N + BN - 1) / BN, (M + BM - 1) / BM);
    gemm_f32_wmma<<<grid, dim3(128), 0, stream>>>(X, W, bias, res, Y, M, N, K, relu);
  };
  auto ln = [&](const float* in, const float* g, const float* b,
                const float* add, float* out) {
    layernorm_k<<<dim3(T_ / 8), dim3(256), 0, stream>>>(in, g, b, add, out, T_);
  };

  // ---- 1. across-query self attention ----
  gemm(tgt, across_in_w, across_in_b, nullptr, buf_h, T_, 768, 256, 0);
  attn_softmax<<<dim3(BS_ * NK_, H_), dim3(128), 0, stream>>>(
      buf_h, buf_a, NQ_, BS_ * NK_, 1, 0.f);
  gemm(buf_a, across_out_w, across_out_b, tgt, buf_b, T_, 256, 256, 0);
  ln(buf_b, across_norm_g, across_norm_b, pos, buf_x);   // + query_pos fused

  // ---- 2. multi-scale deformable attention ----
  gemm(mem, val_w, val_b, nullptr, buf_val, LIN_ * BS_, 256, 256, 0);
  gemm(buf_x, off_w, off_b, nullptr, buf_off, T_, 256, 256, 0);
  gemm(buf_x, aw_w, aw_b, nullptr, buf_aw, T_, 128, 256, 0);
  msdeform_sample<<<dim3(BS_ * NQ_ * NK_), dim3(256), 0, stream>>>(
      buf_off, buf_aw, refp, buf_val, buf_a);
  gemm(buf_a, msout_w, msout_b, buf_x, buf_b, T_, 256, 256, 0);
  ln(buf_b, norm1_g, norm1_b, nullptr, buf_x);

  // ---- 3. FFN ----
  gemm(buf_x, lin1_w, lin1_b, nullptr, buf_h, T_, 1024, 256, 1);
  gemm(buf_h, lin2_w, lin2_b, buf_x, buf_b, T_, 256, 1024, 0);
  ln(buf_b, norm2_g, norm2_b, pos, buf_x);               // + query_pos fused

  // ---- 4. within-key self attention (S=18, mask = +1e-8) ----
  gemm(buf_x, within_in_w, within_in_b, nullptr, buf_h, T_, 768, 256, 0);
  attn_softmax<<<dim3(NQ_ * BS_, H_), dim3(128), 0, stream>>>(
      buf_h, buf_a, NK_, 1, NK_, 1e-8f);
  gemm(buf_a, within_out_w, within_out_b, buf_x, buf_b, T_, 256, 256, 0);
  ln(buf_b, within_norm_g, within_norm_b, nullptr, buf_x);

  // ---- 5. FFN' + final norm -> output ----
  gemm(buf_x, lin1p_w, lin1p_b, nullptr, buf_h, T_, 1024, 256, 1);
  gemm(buf_h, lin2p_w, lin2p_b, buf_x, buf_b, T_, 256, 1024, 0);
  ln(buf_b, norm2p_g, norm2p_b, nullptr, (float*)d_out);
}